// MinibatchDiscrimination_52879637348745
// MI455X (gfx1250) — compile-verified
//
#include <hip/hip_runtime.h>

// Problem sizes (fixed by the reference).
#define B_SZ  1024
#define F_SZ  2048
#define NK    32
#define KD    8
#define MDIM  (NK * KD)        // 256 columns of m
#define OUTW  (F_SZ + NK)      // 2080 output columns

typedef __attribute__((ext_vector_type(2))) float v2f;
typedef __attribute__((ext_vector_type(8))) float v8f;

// ---------------------------------------------------------------------------
// Kernel 1: m = x @ W using V_WMMA_F32_16X16X4_F32.
// One wave computes FOUR 16x16 M-tiles at a fixed N-tile, so each B fragment
// (two strided b32 loads) is reused by 4 independent WMMAs into 4 separate
// accumulators (breaks the serial acc chain seen in round-1 disasm).
// Grid: 16 M-groups x 16 N-tiles = 256 waves, 8 waves/block -> 32 blocks.
// ---------------------------------------------------------------------------
__global__ __launch_bounds__(256) void mb_gemm_wmma_f32(
    const float* __restrict__ X,   // [1024, 2048]
    const float* __restrict__ W,   // [2048, 256]
    float* __restrict__ M)         // [1024, 256]
{
    const int wave  = threadIdx.x >> 5;
    const int lane  = threadIdx.x & 31;
    const int wtile = blockIdx.x * 8 + wave;   // 0..255
    const int mgrp  = wtile >> 4;              // 0..15 -> rows mgrp*64 .. +63
    const int ntile = wtile & 15;              // 0..15

    // A-fragment addressing (16x4 f32): lanes 0-15 hold K={0,1}, 16-31 K={2,3}
    const int row = lane & 15;
    const int kb  = (lane >> 4) << 1;          // 0 or 2 (even -> v2f aligned)

    const float* arow0 = X + (size_t)(mgrp * 64 + row) * F_SZ;
    // B-fragment (4x16 f32): column = lane&15, same K split as A.
    const float* bcol  = W + (ntile * 16 + row);

    v8f acc0 = {}, acc1 = {}, acc2 = {}, acc3 = {};
    for (int k = 0; k < F_SZ; k += 16) {
#pragma unroll
        for (int u = 0; u < 4; ++u) {
            const int kk = k + 4 * u + kb;
            v2f b;
            b.x = bcol[(size_t)kk * MDIM];
            b.y = bcol[(size_t)(kk + 1) * MDIM];
            const v2f a0 = *(const v2f*)(arow0 + kk);                  // b64
            const v2f a1 = *(const v2f*)(arow0 + 16 * F_SZ + kk);
            const v2f a2 = *(const v2f*)(arow0 + 32 * F_SZ + kk);
            const v2f a3 = *(const v2f*)(arow0 + 48 * F_SZ + kk);
            // (neg_a, A, neg_b, B, c_mod, C, reuse_a, reuse_b)
            acc0 = __builtin_amdgcn_wmma_f32_16x16x4_f32(false, a0, false, b, (short)0, acc0, false, false);
            acc1 = __builtin_amdgcn_wmma_f32_16x16x4_f32(false, a1, false, b, (short)0, acc1, false, false);
            acc2 = __builtin_amdgcn_wmma_f32_16x16x4_f32(false, a2, false, b, (short)0, acc2, false, false);
            acc3 = __builtin_amdgcn_wmma_f32_16x16x4_f32(false, a3, false, b, (short)0, acc3, false, false);
        }
    }

    // C/D layout: VGPR r -> M=r (lanes 0-15) / M=8+r (lanes 16-31), N = lane&15
    const int n    = lane & 15;
    const int mofs = (lane >> 4) * 8;
    float* dst = M + (size_t)(mgrp * 64) * MDIM + ntile * 16;
#pragma unroll
    for (int r = 0; r < 8; ++r) {
        dst[(size_t)(mofs + r) * MDIM + n]            = acc0[r];
        dst[(size_t)(16 + mofs + r) * MDIM + n]       = acc1[r];
        dst[(size_t)(32 + mofs + r) * MDIM + n]       = acc2[r];
        dst[(size_t)(48 + mofs + r) * MDIM + n]       = acc3[r];
    }
}

// ---------------------------------------------------------------------------
// Kernel 2: mb_feats[i,k] = sum_j exp(-sum_d |m[i,k,d]-m[j,k,d]|)
// Block = 256 threads: thread t -> (i_local = t>>5 in 0..7, k = t&31).
// j loops over LDS tiles of 32 rows (32 KB). All LDS reads are b128.
// ---------------------------------------------------------------------------
#define TJ 32
__global__ __launch_bounds__(256) void mb_pairwise(
    const float* __restrict__ M,   // [1024, 256]
    float* __restrict__ OUT)       // [1024, 2080]
{
    __shared__ float tile[TJ * MDIM];      // 32 KB

    const int t  = threadIdx.x;
    const int k  = t & 31;
    const int il = t >> 5;
    const int i  = blockIdx.x * 8 + il;

    // This thread's own row m[i,k,:] (8 floats) in registers.
    const float4* mrow = (const float4*)(M + (size_t)i * MDIM + k * KD);
    const float4 mi0 = mrow[0];
    const float4 mi1 = mrow[1];

    float acc = 0.0f;
    for (int j0 = 0; j0 < B_SZ; j0 += TJ) {
        // Cooperative load of 32 rows (8192 floats = 2048 float4, 8/thread).
        const float4* src = (const float4*)(M + (size_t)j0 * MDIM);
        float4* dst = (float4*)tile;
#pragma unroll
        for (int v = 0; v < 8; ++v) {
            const int idx = v * 256 + t;
            dst[idx] = src[idx];
        }
        if (j0 + TJ < B_SZ)  // global_prefetch_b8 of next tile
            __builtin_prefetch(M + (size_t)(j0 + TJ) * MDIM + t * 32, 0, 1);
        __syncthreads();

#pragma unroll 4
        for (int jj = 0; jj < TJ; ++jj) {
            const float4* mj = (const float4*)(tile + jj * MDIM + k * KD);
            const float4 a = mj[0];       // ds_load_b128
            const float4 b = mj[1];       // ds_load_b128
            float dist = fabsf(mi0.x - a.x) + fabsf(mi0.y - a.y)
                       + fabsf(mi0.z - a.z) + fabsf(mi0.w - a.w)
                       + fabsf(mi1.x - b.x) + fabsf(mi1.y - b.y)
                       + fabsf(mi1.z - b.z) + fabsf(mi1.w - b.w);
            acc += exp2f(dist * -1.44269504088896340736f);  // exp(-dist)
        }
        __syncthreads();
    }

    OUT[(size_t)i * OUTW + F_SZ + k] = acc;
}

// ---------------------------------------------------------------------------
// Kernel 3: copy x into out[:, 0:2048] (float4 streaming; both sides 16B
// aligned since 2080*4 bytes is a multiple of 16).
// ---------------------------------------------------------------------------
__global__ __launch_bounds__(256) void mb_copy_x(
    const float4* __restrict__ X4, float* __restrict__ OUT)
{
    const int t   = blockIdx.x * 256 + threadIdx.x;  // 0 .. 1024*512-1
    const int row = t >> 9;                          // 512 float4 per row
    const int c   = t & 511;
    ((float4*)(OUT + (size_t)row * OUTW))[c] = X4[(size_t)row * 512 + c];
}

// ---------------------------------------------------------------------------
extern "C" void kernel_launch(void* const* d_in, const int* in_sizes, int n_in,
                              void* d_out, int out_size, void* d_ws, size_t ws_size,
                              hipStream_t stream) {
    const float* x = (const float*)d_in[0];   // [1024, 2048] f32
    const float* W = (const float*)d_in[1];   // [2048, 256]  f32
    float* out = (float*)d_out;               // [1024, 2080] f32
    float* m   = (float*)d_ws;                // scratch: 1024*256*4 = 1 MB

    mb_gemm_wmma_f32<<<32, 256, 0, stream>>>(x, W, m);
    mb_copy_x<<<(B_SZ * (F_SZ / 4)) / 256, 256, 0, stream>>>((const float4*)x, out);
    mb_pairwise<<<B_SZ / 8, 256, 0, stream>>>(m, out);
}